// DynamicExpertGating_15496242004075
// MI455X (gfx1250) — compile-verified
//
#include <hip/hip_runtime.h>
#include <math.h>

#define Bz 4
#define Sz 2048
#define Hh 1024
#define NE 8
#define ED 1024
#define NTOK (Bz * Sz)       /* 8192 tokens  */
#define NSLOT (NTOK * 2)     /* 16384 (token,k) slots */
#define CAP NSLOT            /* worst case: all slots on one expert */
#define OUT_ELEMS ((size_t)NSLOT * Hh)

typedef __attribute__((ext_vector_type(2))) float v2f;
typedef __attribute__((ext_vector_type(8))) float v8f;

// Native CDNA5 transcendental tanh (VOP1 trans op). ISA hazard rule: 1
// independent op / V_NOP after a trans op before its result is consumed —
// the compiler can't see through the asm, so pad with v_nop explicitly.
__device__ __forceinline__ float fast_tanh(float v) {
  float r;
  asm volatile("v_tanh_f32 %0, %1\n\tv_nop\n\tv_nop" : "=v"(r) : "v"(v));
  return r;
}

__device__ __forceinline__ float gelu_tanh(float v) {
  const float c = 0.7978845608028654f;
  float t = fast_tanh(c * (v + 0.044715f * v * v * v));
  return 0.5f * v * (1.0f + t);
}

// ---------------------------------------------------------------- init
__global__ void moe_init(int* counts) {
  if (threadIdx.x < NE) counts[threadIdx.x] = 0;
}

// ---------------------------------------------------------------- router
// one wave (32 lanes) per token; 8 waves / block
__global__ __launch_bounds__(256) void moe_router(
    const float* __restrict__ x, const float* __restrict__ rw,
    const float* __restrict__ rb, int* __restrict__ counts,
    int* __restrict__ slotlist, float* __restrict__ pnorm,
    float* __restrict__ lsq) {
  const int lane = threadIdx.x & 31;
  const int token = blockIdx.x * 8 + (threadIdx.x >> 5);
  if (token >= NTOK) return;
  const float* xr = x + (size_t)token * Hh;
  float acc[NE];
#pragma unroll
  for (int e = 0; e < NE; ++e) acc[e] = 0.0f;
  for (int j = lane; j < Hh; j += 32) {
    float xv = xr[j];
    const float* w = rw + (size_t)j * NE;
#pragma unroll
    for (int e = 0; e < NE; ++e) acc[e] += xv * w[e];
  }
#pragma unroll
  for (int e = 0; e < NE; ++e) {
#pragma unroll
    for (int off = 16; off > 0; off >>= 1)
      acc[e] += __shfl_xor(acc[e], off, 32);
  }
  if (lane == 0) {
    float lg[NE];
    float m = -3.4e38f;
#pragma unroll
    for (int e = 0; e < NE; ++e) {
      lg[e] = acc[e] + rb[e];
      m = fmaxf(m, lg[e]);
    }
    float se = 0.0f;
#pragma unroll
    for (int e = 0; e < NE; ++e) se += __expf(lg[e] - m);
    float lse = m + __logf(se);
    lsq[token] = lse * lse;  // deterministic reduce later
    // top-2 (ties -> lowest index, matching lax.top_k)
    int i1 = 0;
#pragma unroll
    for (int e = 1; e < NE; ++e)
      if (lg[e] > lg[i1]) i1 = e;
    int i2 = (i1 == 0) ? 1 : 0;
#pragma unroll
    for (int e = 0; e < NE; ++e)
      if (e != i1 && lg[e] > lg[i2]) i2 = e;
    float p1 = __expf(lg[i1] - m) / se;
    float p2 = __expf(lg[i2] - m) / se;
    float inv = 1.0f / (p1 + p2);
    int s0 = token * 2, s1 = s0 + 1;
    pnorm[s0] = p1 * inv;
    pnorm[s1] = p2 * inv;
    int pos = atomicAdd(&counts[i1], 1);
    slotlist[i1 * CAP + pos] = s0;
    pos = atomicAdd(&counts[i2], 1);
    slotlist[i2 * CAP + pos] = s1;
  }
}

// ---------------------------------------------------------------- z-loss
__global__ void moe_zreduce(const float* __restrict__ lsq, float* zslot) {
  __shared__ float sm[256];
  float a = 0.0f;
  for (int i = threadIdx.x; i < NTOK; i += 256) a += lsq[i];
  sm[threadIdx.x] = a;
  __syncthreads();
  for (int s = 128; s > 0; s >>= 1) {
    if ((int)threadIdx.x < s) sm[threadIdx.x] += sm[threadIdx.x + s];
    __syncthreads();
  }
  if (threadIdx.x == 0) *zslot = sm[0] * (0.001f / (float)NTOK);
}

// ---------------------------------------------------------------- fused FFN
// grid = (CAP/16, NE); block = 256 (8 wave32). Each block: 16 gathered slot
// rows x 1024 output cols, fused  gelu(x*wi)  ->  *wo , h kept in LDS.
// x tiles are staged with double-buffered GLOBAL_LOAD_ASYNC_TO_LDS_B128
// (ASYNCcnt): each wave issues exactly one async instr per 16x64 chunk, so
// s_wait_asynccnt 0x1 == "oldest chunk landed".
__global__ __launch_bounds__(256) void moe_ffn(
    const float* __restrict__ x, const float* __restrict__ wi_w,
    const float* __restrict__ wi_b, const float* __restrict__ wo_w,
    const float* __restrict__ wo_b, const int* __restrict__ counts,
    const int* __restrict__ slotlist, const float* __restrict__ pnorm,
    float* __restrict__ out) {
  const int e = blockIdx.y;
  const int n = counts[e];
  const int rowBase = blockIdx.x * 16;
  if (rowBase >= n) return;

  __shared__ int sSlot[16];
  __shared__ int sTok[16];
  __shared__ float sA[2][16][65];  // double-buffered x chunk 16 x 64 (+pad)
  __shared__ float sH[16][513];    // h half 16 x 512 (+pad)

  const int tid = threadIdx.x;
  if (tid < 16) {
    int r = rowBase + tid;
    int slot = slotlist[e * CAP + (r < n ? r : n - 1)];
    sSlot[tid] = slot;
    sTok[tid] = slot >> 1;  // slot = token*2 + k
  }
  __syncthreads();

  const int wave = tid >> 5;
  const int lane = tid & 31;
  const int lhi = lane >> 4;  // lane group (0: lanes 0-15, 1: 16-31)
  const int lm = lane & 15;
  const float* wiW = wi_w + (size_t)e * Hh * ED;
  const float* woW = wo_w + (size_t)e * ED * ED;

  // async stage of one 16x64 x-chunk: 256 threads x 16B
  const int ar = tid >> 4;          // row 0..15
  const int ac = (tid & 15) << 2;   // col 0..60 step 4
  const size_t axbase = (size_t)sTok[ar] * Hh + ac;

  v8f c2[8];
#pragma unroll
  for (int t = 0; t < 8; ++t) c2[t] = (v8f)(0.0f);

  for (int ph = 0; ph < 2; ++ph) {  // h columns [ph*512, ph*512+512)
    v8f c1[4];
#pragma unroll
    for (int t = 0; t < 4; ++t) c1[t] = (v8f)(0.0f);

    // prologue: issue chunk kc=0 into buffer 0
    {
      const float* g = x + axbase;
      unsigned l = (unsigned)(size_t)&sA[0][ar][ac];
      asm volatile("global_load_async_to_lds_b128 %0, %1, off"
                   :: "v"(l), "v"(g) : "memory");
    }

    // ---- stage 1: h = x * wi  (K over full 1024, staged 64 at a time)
    for (int kc = 0; kc < Hh; kc += 64) {
      const int buf = (kc >> 6) & 1;
      if (kc + 64 < Hh) {  // issue next chunk into the other buffer
        const float* g = x + axbase + kc + 64;
        unsigned l = (unsigned)(size_t)&sA[buf ^ 1][ar][ac];
        asm volatile("global_load_async_to_lds_b128 %0, %1, off"
                     :: "v"(l), "v"(g) : "memory");
        asm volatile("s_wait_asynccnt 0x1" ::: "memory");
      } else {
        asm volatile("s_wait_asynccnt 0x0" ::: "memory");
      }
      __syncthreads();
#pragma unroll 2
      for (int kk = 0; kk < 64; kk += 4) {
        // A 16x4 f32: VGPR v -> K = v + 2*lhi  (ISA 7.12.2)
        int ka = kk + 2 * lhi;
        v2f a = {sA[buf][lm][ka], sA[buf][lm][ka + 1]};
        int kb = kc + kk + 2 * lhi;
#pragma unroll
        for (int t = 0; t < 4; ++t) {
          int col = ph * 512 + wave * 64 + t * 16 + lm;
          v2f b = {wiW[(size_t)kb * ED + col], wiW[(size_t)(kb + 1) * ED + col]};
          c1[t] = __builtin_amdgcn_wmma_f32_16x16x4_f32(
              false, a, false, b, (short)0, c1[t], false, false);
        }
      }
      __syncthreads();  // all waves done with buf before it is re-filled
    }
    // ---- bias + gelu, spill C-frag to LDS (C layout: M = r + 8*lhi, N = lm)
#pragma unroll
    for (int t = 0; t < 4; ++t) {
      int lcol = wave * 64 + t * 16 + lm;
      float bias = wi_b[e * ED + ph * 512 + lcol];
#pragma unroll
      for (int r = 0; r < 8; ++r) {
        int m = r + 8 * lhi;
        sH[m][lcol] = gelu_tanh(c1[t][r] + bias);
      }
    }
    __syncthreads();

    // ---- stage 2: out += h * wo  (K over this phase's 512 h-cols)
#pragma unroll 2
    for (int kk = 0; kk < 512; kk += 4) {
      int ka = kk + 2 * lhi;
      v2f a = {sH[lm][ka], sH[lm][ka + 1]};
      int kb = ph * 512 + kk + 2 * lhi;
#pragma unroll
      for (int t = 0; t < 8; ++t) {
        int col = wave * 128 + t * 16 + lm;
        v2f b = {woW[(size_t)kb * ED + col], woW[(size_t)(kb + 1) * ED + col]};
        c2[t] = __builtin_amdgcn_wmma_f32_16x16x4_f32(
            false, a, false, b, (short)0, c2[t], false, false);
      }
    }
    __syncthreads();  // before next phase overwrites sH
  }

  // ---- epilogue: bias, scale by routing prob, streamed (NT) store
#pragma unroll
  for (int t = 0; t < 8; ++t) {
    int col = wave * 128 + t * 16 + lm;
    float bias = wo_b[e * ED + col];
#pragma unroll
    for (int r = 0; r < 8; ++r) {
      int m = r + 8 * lhi;
      if (rowBase + m < n) {
        int slot = sSlot[m];
        float v = (c2[t][r] + bias) * pnorm[slot];
        __builtin_nontemporal_store(v, &out[(size_t)slot * ED + col]);
      }
    }
  }
}

// ---------------------------------------------------------------- launch
extern "C" void kernel_launch(void* const* d_in, const int* in_sizes, int n_in,
                              void* d_out, int out_size, void* d_ws,
                              size_t ws_size, hipStream_t stream) {
  const float* x    = (const float*)d_in[0];
  const float* rw   = (const float*)d_in[1];
  const float* rb   = (const float*)d_in[2];
  const float* wi_w = (const float*)d_in[3];
  const float* wi_b = (const float*)d_in[4];
  const float* wo_w = (const float*)d_in[5];
  const float* wo_b = (const float*)d_in[6];
  float* out = (float*)d_out;
  float* zslot = out + OUT_ELEMS;

  char* ws = (char*)d_ws;
  int* counts = (int*)ws;                                      // 8 ints
  int* slotlist = (int*)(ws + 256);                            // 8*16384 ints
  float* pnorm = (float*)(ws + 256 + sizeof(int) * NE * CAP);  // 16384 f32
  float* lsq = pnorm + NSLOT;                                  // 8192 f32

  moe_init<<<1, 64, 0, stream>>>(counts);
  moe_router<<<NTOK / 8, 256, 0, stream>>>(x, rw, rb, counts, slotlist, pnorm,
                                           lsq);
  moe_zreduce<<<1, 256, 0, stream>>>(lsq, zslot);
  dim3 g(CAP / 16, NE);
  moe_ffn<<<g, 256, 0, stream>>>(x, wi_w, wi_b, wo_w, wo_b, counts, slotlist,
                                 pnorm, out);
}